// SubgraphGNN_90134183673868
// MI455X (gfx1250) — compile-verified
//
#include <hip/hip_runtime.h>
#include <math.h>

// ---- problem constants ----
#define NN 30000
#define EE 480000
#define DD 128
#define RR 16
#define HH 4
#define DHH 32
#define LL 3

// LDS row stride (floats) for both A (row-major, 128 k's) and Bt (transposed,
// 128 k's contiguous per n-row): 132 % 64 == 4 makes both WMMA fragment read
// patterns (16 rows x consecutive-pair) hit 64 distinct banks. Even => b64 ok.
#define PK 132

typedef __attribute__((ext_vector_type(2))) float v2f;
typedef __attribute__((ext_vector_type(8))) float v8f;
typedef __attribute__((ext_vector_type(4))) unsigned int v4u;
typedef __attribute__((ext_vector_type(4))) int v4i;
typedef __attribute__((ext_vector_type(8))) int v8i;

// LDS byte offset of a __shared__ object (addrspacecast generic -> AS3).
__device__ __forceinline__ unsigned lds_off_of(const void* p) {
  return (unsigned)(unsigned long long)(__attribute__((address_space(3))) const char*)p;
}

// Tensor Data Mover: 2D tile (rows of 128 floats) -> LDS with 4-DWORD pad per
// 128 DWORDs (row stride 132 floats); optional 16-bit-index gather mode.
__device__ __forceinline__ void tdm_load_2d(unsigned lds_off, const void* gptr,
                                            unsigned tensor_rows, unsigned tile_rows,
                                            unsigned gather16, v4i g2, v4i g3) {
  unsigned long long ga = (unsigned long long)gptr;
  v4u g0;
  g0.x = 1u | (gather16 << 31);                                  // count=1 | gather_mode
  g0.y = lds_off;                                                // LDS byte address
  g0.z = (unsigned)(ga & 0xFFFFFFFFull);                         // global addr lo
  g0.w = (unsigned)((ga >> 32) & 0x01FFFFFFull) | (2u << 30);    // addr hi | type=2
  v8i g1;
  g1[0] = (int)((2u << 16) | (1u << 20) |                        // data_size=4B | pad_en
                (6u << 22) | (3u << 25));                        // pad 4 DW per 128 DW
  g1[1] = (int)(((unsigned)DD & 0xFFFFu) << 16);                 // tensor_dim0=128 (lo)
  g1[2] = (int)((tensor_rows & 0xFFFFu) << 16);                  // dim0 hi=0 | dim1 lo
  g1[3] = (int)(((tensor_rows >> 16) & 0xFFFFu) | ((unsigned)DD << 16)); // dim1 hi | tile_dim0
  g1[4] = (int)(tile_rows & 0xFFFFu);                            // tile_dim1 | tile_dim2=0
  g1[5] = (int)DD;                                               // tensor_dim0_stride=128
  g1[6] = 0;
  g1[7] = 0;
  v8i g4 = {};                                                   // unused 5th SGPR group
  __builtin_amdgcn_tensor_load_to_lds(g0, g1, g2, g3, g4, 0);
}

// ---------------- utility kernels ----------------
__global__ void fill_f32_kernel(float* p, float v, int n) {
  int i = blockIdx.x * blockDim.x + threadIdx.x;
  if (i < n) p[i] = v;
}
__global__ void fill_i32_kernel(int* p, int v, int n) {
  int i = blockIdx.x * blockDim.x + threadIdx.x;
  if (i < n) p[i] = v;
}
__global__ void reduce_sum_kernel(const float* __restrict__ x, int n, float* acc) {
  int stride = gridDim.x * blockDim.x;
  float s = 0.f;
  for (int i = blockIdx.x * blockDim.x + threadIdx.x; i < n; i += stride) s += x[i];
  atomicAdd(acc, s);
}

// Transpose a batch of contiguous 128x128 matrices: out[m][n*128+k]=in[m][k*128+n]
__global__ void transpose128_kernel(const float* __restrict__ in, float* __restrict__ out) {
  const float* ip = in + (size_t)blockIdx.x * (DD * DD);
  float* op = out + (size_t)blockIdx.x * (DD * DD);
  for (int idx = threadIdx.x; idx < DD * DD; idx += blockDim.x) {
    int k = idx >> 7, n = idx & 127;
    op[n * DD + k] = ip[idx];
  }
}

// ---------------- edge sort by etype (counting sort) ----------------
__global__ void hist_kernel(const int* __restrict__ etype, int* cnt, int n) {
  int i = blockIdx.x * blockDim.x + threadIdx.x;
  if (i < n) atomicAdd(&cnt[etype[i]], 1);
}
__global__ void scan16_kernel(const int* __restrict__ cnt, int* cur) {
  if (threadIdx.x == 0) {
    int s = 0;
    for (int r = 0; r < RR; ++r) { cur[r] = s; s += cnt[r]; }
  }
}
__global__ void sort_kernel(const int* __restrict__ etype, int* cur, int* eperm, int n) {
  int i = blockIdx.x * blockDim.x + threadIdx.x;
  if (i < n) {
    int t = etype[i];
    int p = atomicAdd(&cur[t], 1);
    eperm[p] = i;
  }
}

// ---------------- RelGraphConv: TDM-fed WMMA edge GEMM + scatter ----------
// Block = 64 sorted edges (4 waves x 16). A rows gathered by TDM gather-mode;
// transposed W_rel[r] staged to LDS by 4 quarter TDM loads; per-run
// 16x128x128 WMMA GEMM; float-atomic scatter to h_rel[dst].
__global__ void __launch_bounds__(128)
rel_edge_gemm_kernel(const float* __restrict__ h, const float* __restrict__ Wtl,
                     const int* __restrict__ src, const int* __restrict__ dst,
                     const int* __restrict__ etype, const int* __restrict__ eperm,
                     float* __restrict__ hrel, int nedges) {
  __shared__ float sBt[DD * PK];    // 67,584 B (transposed weights: [n][k])
  __shared__ float sA[64 * PK];     // 33,792 B
  __shared__ int sDst[64];
  __shared__ int sEt[64];
  __shared__ int sSrc[64];

  int te = blockIdx.x * 64;
  int t = threadIdx.x;
  if (t < 64) {
    int ei = te + t;
    if (ei < nedges) {
      int e = eperm[ei];
      sSrc[t] = src[e]; sDst[t] = dst[e]; sEt[t] = etype[e];
    } else { sSrc[t] = 0; sDst[t] = -1; sEt[t] = -1; }
  }
  __syncthreads();

  int lane = t & 31, wave = t >> 5;
  // gather-mode TDM: this wave's 16 h[src] rows -> its A tile
  {
    unsigned idx[16];
    #pragma unroll
    for (int q = 0; q < 16; ++q) idx[q] = (unsigned)sSrc[wave * 16 + q] & 0xFFFFu;
    v4i g2, g3;
    g2[0] = (int)(idx[0] | (idx[1] << 16));
    g2[1] = (int)(idx[2] | (idx[3] << 16));
    g2[2] = (int)(idx[4] | (idx[5] << 16));
    g2[3] = (int)(idx[6] | (idx[7] << 16));
    g3[0] = (int)(idx[8] | (idx[9] << 16));
    g3[1] = (int)(idx[10] | (idx[11] << 16));
    g3[2] = (int)(idx[12] | (idx[13] << 16));
    g3[3] = (int)(idx[14] | (idx[15] << 16));
    tdm_load_2d(lds_off_of(sA) + (unsigned)(wave * 16 * PK * 4), h,
                (unsigned)NN, 16u, 1u, g2, g3);
  }

  int ma = lane & 15, half = lane >> 4, kb = half * 2, nn = lane & 15;
  int i = 0;
  while (i < 64) {                    // uniform run loop over equal-etype runs
    int r = sEt[i];
    if (r < 0) { ++i; continue; }
    int j = i + 1;
    while (j < 64 && sEt[j] == r) ++j;

    __syncthreads();                  // previous run's sBt reads retired
    {
      v4i z = {};
      tdm_load_2d(lds_off_of(sBt) + (unsigned)(wave * 32 * PK * 4),
                  Wtl + (size_t)r * (DD * DD) + (size_t)wave * 32 * DD,
                  32u, 32u, 0u, z, z);
    }
    __builtin_amdgcn_s_wait_tensorcnt(0);
    __syncthreads();                  // all quarters (and A on first run) in LDS

    int lo = i - wave * 16; if (lo < 0) lo = 0;
    int hi = j - wave * 16; if (hi > 16) hi = 16;
    if (lo < hi) {                    // wave-uniform
      v8f c[8] = {};
      const int arow = (wave * 16 + ma) * PK;
      #pragma unroll 4
      for (int k = 0; k < DD; k += 4) {
        v2f a = *(const v2f*)&sA[arow + k + kb];
        #pragma unroll
        for (int f = 0; f < 8; ++f) {
          v2f b = *(const v2f*)&sBt[(f * 16 + nn) * PK + k + kb];
          c[f] = __builtin_amdgcn_wmma_f32_16x16x4_f32(false, a, false, b, (short)0, c[f], false, false);
        }
      }
      #pragma unroll
      for (int f = 0; f < 8; ++f) {
        #pragma unroll
        for (int v = 0; v < 8; ++v) {
          int mt = v + half * 8;
          if (mt >= lo && mt < hi) {
            int dn = sDst[wave * 16 + mt];
            atomicAdd(&hrel[(size_t)dn * DD + f * 16 + nn], c[f][v]);
          }
        }
      }
    }
    i = j;
  }
}

// ---------------- generic dense WMMA GEMM: out[M,128] (+)= A[M,128]@B[128,128] + bias
// Block = 64 rows (4 waves x 16). A and transposed-B staged to LDS via TDM.
__global__ void __launch_bounds__(128)
dense_gemm_kernel(const float* __restrict__ A, const float* __restrict__ Bt,
                  const float* __restrict__ bias, float* __restrict__ out,
                  int M, int accumulate) {
  __shared__ float sBt[DD * PK];
  __shared__ float sA[64 * PK];
  int m0 = blockIdx.x * 64;
  int t = threadIdx.x;
  int lane = t & 31, wave = t >> 5;

  {
    v4i z = {};
    tdm_load_2d(lds_off_of(sBt) + (unsigned)(wave * 32 * PK * 4),
                Bt + (size_t)wave * 32 * DD, 32u, 32u, 0u, z, z);
    if (wave < 2) {                    // two A halves (32 rows each), M-clamped
      int r0 = m0 + wave * 32;
      int rem = M - r0; if (rem < 0) rem = 0; if (rem > 32) rem = 32;
      tdm_load_2d(lds_off_of(sA) + (unsigned)(wave * 32 * PK * 4),
                  A + (size_t)r0 * DD, (unsigned)rem, 32u, 0u, z, z);
    }
  }
  __builtin_amdgcn_s_wait_tensorcnt(0);
  __syncthreads();

  int ma = lane & 15, half = lane >> 4, kb = half * 2, nn = lane & 15;
  v8f c[8] = {};
  const int arow = (wave * 16 + ma) * PK;
  #pragma unroll 4
  for (int k = 0; k < DD; k += 4) {
    v2f a = *(const v2f*)&sA[arow + k + kb];
    #pragma unroll
    for (int f = 0; f < 8; ++f) {
      v2f b = *(const v2f*)&sBt[(f * 16 + nn) * PK + k + kb];
      c[f] = __builtin_amdgcn_wmma_f32_16x16x4_f32(false, a, false, b, (short)0, c[f], false, false);
    }
  }
  #pragma unroll
  for (int f = 0; f < 8; ++f) {
    #pragma unroll
    for (int v = 0; v < 8; ++v) {
      int mt = v + half * 8;
      int gr = m0 + wave * 16 + mt;
      if (gr < M) {
        int nc = f * 16 + nn;
        float rv = c[f][v];
        if (bias) rv += bias[nc];
        float* o = out + (size_t)gr * DD + nc;
        if (accumulate) *o += rv; else *o = rv;
      }
    }
  }
}

// ---------------- GAT pieces ----------------
__global__ void att_node_kernel(const float* __restrict__ z,
                                const float* __restrict__ al,
                                const float* __restrict__ ar,
                                float* __restrict__ el, float* __restrict__ er, int nh) {
  int idx = blockIdx.x * blockDim.x + threadIdx.x;
  if (idx >= nh) return;
  int node = idx >> 2, hh = idx & 3;
  const float* zp = z + (size_t)node * DD + hh * DHH;
  const float* alp = al + hh * DHH;
  const float* arp = ar + hh * DHH;
  float sl = 0.f, sr = 0.f;
  #pragma unroll
  for (int d = 0; d < DHH; ++d) { float zv = zp[d]; sl += zv * alp[d]; sr += zv * arp[d]; }
  el[idx] = sl; er[idx] = sr;
}

__device__ inline void atomicMaxF(float* addr, float val) {
  if (val >= 0.f) atomicMax((int*)addr, __float_as_int(val));
  else            atomicMin((unsigned int*)addr, __float_as_uint(val));
}

__global__ void edge_max_kernel(const float* __restrict__ el, const float* __restrict__ er,
                                const int* __restrict__ src, const int* __restrict__ dst,
                                float* __restrict__ ebuf, float* __restrict__ mbuf, int neh) {
  int idx = blockIdx.x * blockDim.x + threadIdx.x;
  if (idx >= neh) return;
  int e = idx >> 2, hh = idx & 3;
  float x = el[src[e] * HH + hh] + er[dst[e] * HH + hh];
  x = x > 0.f ? x : 0.2f * x;                 // leaky_relu(0.2)
  ebuf[idx] = x;
  atomicMaxF(&mbuf[dst[e] * HH + hh], x);
}

__global__ void edge_exp_kernel(const int* __restrict__ dst,
                                float* __restrict__ ebuf,
                                const float* __restrict__ mbuf,
                                float* __restrict__ denom, int neh) {
  int idx = blockIdx.x * blockDim.x + threadIdx.x;
  if (idx >= neh) return;
  int e = idx >> 2, hh = idx & 3;
  float ex = expf(ebuf[idx] - mbuf[dst[e] * HH + hh]);
  ebuf[idx] = ex;
  atomicAdd(&denom[dst[e] * HH + hh], ex);
}

__global__ void edge_aggr_kernel(const float* __restrict__ ebuf,
                                 const float* __restrict__ denom,
                                 const float* __restrict__ z,
                                 const int* __restrict__ src, const int* __restrict__ dst,
                                 float* __restrict__ hgat) {
  int e = blockIdx.x;
  int n = threadIdx.x;
  __shared__ float sAlpha[HH];
  __shared__ int sS, sD;
  if (n == 0) { sS = src[e]; sD = dst[e]; }
  __syncthreads();
  if (n < HH) sAlpha[n] = ebuf[e * HH + n] / denom[sD * HH + n];
  __syncthreads();
  float v = sAlpha[n >> 5] * z[(size_t)sS * DD + n];
  atomicAdd(&hgat[(size_t)sD * DD + n], v);
}

// ---------------- bias folding + epilogues ----------------
__global__ void fuse_bias_kernel(const float* __restrict__ b_gat_l,
                                 const float* __restrict__ W_fuse,
                                 const float* __restrict__ b_fuse,
                                 float* __restrict__ biasF) {
  int n = threadIdx.x;
  float s = b_fuse[n];
  for (int k = 0; k < DD; ++k) s += b_gat_l[k] * W_fuse[(size_t)(DD + k) * DD + n];
  biasF[n] = s;
}
__global__ void gate_bias_kernel(const float* __restrict__ twsum,
                                 const float* __restrict__ W_gate,
                                 const float* __restrict__ b_gate,
                                 float* __restrict__ biasG) {
  int n = threadIdx.x;
  float m = twsum[0] * (1.0f / (float)EE);
  biasG[n] = m * W_gate[(size_t)DD * DD + n] + b_gate[n];
}
__global__ void finish_kernel(const float* __restrict__ hc, const float* __restrict__ g,
                              float* __restrict__ hout, int use_gate, int n) {
  int i = blockIdx.x * blockDim.x + threadIdx.x;
  if (i >= n) return;
  float v = hc[i];
  if (use_gate) v *= 1.0f / (1.0f + expf(-g[i]));
  hout[i] = v > 0.f ? v : 0.f;
}

// ---------------- launcher ----------------
extern "C" void kernel_launch(void* const* d_in, const int* in_sizes, int n_in,
                              void* d_out, int out_size, void* d_ws, size_t ws_size,
                              hipStream_t stream) {
  (void)in_sizes; (void)n_in; (void)out_size; (void)ws_size;
  const float* feat   = (const float*)d_in[0];
  const int*   src    = (const int*)d_in[1];
  const int*   dst    = (const int*)d_in[2];
  const int*   etype  = (const int*)d_in[3];
  const float* tw     = (const float*)d_in[4];
  const float* W_rel  = (const float*)d_in[5];   // [L,R,D,D]
  const float* b_rel  = (const float*)d_in[6];   // [L,D]
  const float* W_loop = (const float*)d_in[7];   // [L,D,D]
  const float* W_gat  = (const float*)d_in[8];   // [L,D,D]
  const float* attn_l = (const float*)d_in[9];   // [L,H,DH]
  const float* attn_r = (const float*)d_in[10];
  const float* b_gat  = (const float*)d_in[11];  // [L,D]
  const float* W_fuse = (const float*)d_in[12];  // [2D,D]
  const float* b_fuse = (const float*)d_in[13];
  const float* W_gate = (const float*)d_in[14];  // [D+1,D]
  const float* b_gate = (const float*)d_in[15];
  float* out = (float*)d_out;

  const int ND = NN * DD;
  const int NH = NN * HH;
  const int EH = EE * HH;
  const int MM = DD * DD;   // one 128x128 matrix

  float* w      = (float*)d_ws;
  float* h_rel  = w;            w += ND;
  float* h_gat  = w;            w += ND;
  float* z      = w;            w += ND;
  float* hc     = w;            w += ND;
  float* g      = w;            w += ND;
  float* hA     = w;            w += ND;
  float* hB     = w;            w += ND;
  float* el     = w;            w += NH;
  float* er     = w;            w += NH;
  float* mbuf   = w;            w += NH;
  float* denom  = w;            w += NH;
  float* ebuf   = w;            w += EH;
  float* wt_rel = w;            w += (size_t)LL * RR * MM;  // transposed weights
  float* wt_loop= w;            w += (size_t)LL * MM;
  float* wt_gat = w;            w += (size_t)LL * MM;
  float* wt_fuse= w;            w += 2 * MM;
  float* wt_gate= w;            w += MM;
  int*   eperm  = (int*)w;
  int*   cnt    = eperm + EE;
  int*   cur    = cnt + RR;
  float* twsum  = (float*)(cur + RR);
  float* biasF  = twsum + 1;
  float* biasG  = biasF + DD;

  // ---- per launch: weight transposes, temporal mean, edge counting-sort ----
  transpose128_kernel<<<LL * RR, 256, 0, stream>>>(W_rel, wt_rel);
  transpose128_kernel<<<LL, 256, 0, stream>>>(W_loop, wt_loop);
  transpose128_kernel<<<LL, 256, 0, stream>>>(W_gat, wt_gat);
  transpose128_kernel<<<2, 256, 0, stream>>>(W_fuse, wt_fuse);
  transpose128_kernel<<<1, 256, 0, stream>>>(W_gate, wt_gate);

  fill_f32_kernel<<<1, 32, 0, stream>>>(twsum, 0.f, 1);
  fill_i32_kernel<<<1, 32, 0, stream>>>(cnt, 0, RR);
  reduce_sum_kernel<<<256, 256, 0, stream>>>(tw, EE, twsum);
  hist_kernel<<<(EE + 255) / 256, 256, 0, stream>>>(etype, cnt, EE);
  scan16_kernel<<<1, 32, 0, stream>>>(cnt, cur);
  sort_kernel<<<(EE + 255) / 256, 256, 0, stream>>>(etype, cur, eperm, EE);

  const float* hin = feat;
  for (int l = 0; l < LL; ++l) {
    float* hout = (l == 0) ? hA : (l == 1) ? hB : out;

    fill_f32_kernel<<<(ND + 255) / 256, 256, 0, stream>>>(h_rel, 0.f, ND);
    fill_f32_kernel<<<(ND + 255) / 256, 256, 0, stream>>>(h_gat, 0.f, ND);
    fill_f32_kernel<<<(NH + 255) / 256, 256, 0, stream>>>(mbuf, -INFINITY, NH);
    fill_f32_kernel<<<(NH + 255) / 256, 256, 0, stream>>>(denom, 0.f, NH);

    // RelGraphConv: TDM+WMMA edge transform + scatter, then self-loop GEMM
    rel_edge_gemm_kernel<<<(EE + 63) / 64, 128, 0, stream>>>(
        hin, wt_rel + (size_t)l * RR * MM, src, dst, etype, eperm, h_rel, EE);
    dense_gemm_kernel<<<(NN + 63) / 64, 128, 0, stream>>>(
        hin, wt_loop + (size_t)l * MM, b_rel + (size_t)l * DD, h_rel, NN, 1);

    // GATConv
    dense_gemm_kernel<<<(NN + 63) / 64, 128, 0, stream>>>(
        hin, wt_gat + (size_t)l * MM, nullptr, z, NN, 0);
    att_node_kernel<<<(NH + 255) / 256, 256, 0, stream>>>(
        z, attn_l + (size_t)l * HH * DHH, attn_r + (size_t)l * HH * DHH, el, er, NH);
    edge_max_kernel<<<(EH + 255) / 256, 256, 0, stream>>>(el, er, src, dst, ebuf, mbuf, EH);
    edge_exp_kernel<<<(EH + 255) / 256, 256, 0, stream>>>(dst, ebuf, mbuf, denom, EH);
    edge_aggr_kernel<<<EE, 128, 0, stream>>>(ebuf, denom, z, src, dst, h_gat);

    // fuse: hc = [h_rel | h_gat] @ W_fuse + b_fuse (b_gat folded into bias)
    fuse_bias_kernel<<<1, 128, 0, stream>>>(b_gat + (size_t)l * DD, W_fuse, b_fuse, biasF);
    dense_gemm_kernel<<<(NN + 63) / 64, 128, 0, stream>>>(h_rel, wt_fuse, biasF, hc, NN, 0);
    dense_gemm_kernel<<<(NN + 63) / 64, 128, 0, stream>>>(h_gat, wt_fuse + MM, nullptr, hc, NN, 1);

    if (l == 0) {
      gate_bias_kernel<<<1, 128, 0, stream>>>(twsum, W_gate, b_gate, biasG);
      dense_gemm_kernel<<<(NN + 63) / 64, 128, 0, stream>>>(hc, wt_gate, biasG, g, NN, 0);
    }
    finish_kernel<<<(ND + 255) / 256, 256, 0, stream>>>(hc, g, hout, l == 0 ? 1 : 0, ND);
    hin = hout;
  }
}